// SymmetricContraction_5772436046498
// MI455X (gfx1250) — compile-verified
//
#include <hip/hip_runtime.h>

// ---------------------------------------------------------------------------
// MACE SymmetricContraction on gfx1250 via V_WMMA_F32_16X16X4_F32.
//
// Degree-3 + degree-2 terms recast as a shared-basis GEMM, computed
// TRANSPOSED so the D-fragment layout matches the epilogue:
//   C2T[(t,k,i) tile: j, r] = sum_m  Bbasis[j, m] * Y[r, m]
//     A operand (16x4, M=j): packed basis fragments (global, L2-hot)
//     B operand (4x16, N=r): per-row Y vector from LDS
//   m in [0,176):  Y = x[r,l] * W3[e,p,c],  basis = U3[k,i,j,l,p] (m = l*11+p)
//   m in [176,180): Y = W2[e,p,c],          basis = U2[k,i,j,p]
// Epilogue per tile: lane holds row r = lane%16, regs hold j = v + 8*half ->
// j-contraction = 8 register FMAs + one shfl_xor(16); Horner over i in one
// scalar register with the U1*W1 degree-1 term folded in.
// ---------------------------------------------------------------------------

typedef __attribute__((ext_vector_type(2))) float v2f;
typedef __attribute__((ext_vector_type(8))) float v8f;

#define FEAT 16
#define KDIM 180        // GEMM inner dimension (176 + 4)
#define KSTEPS 45       // 180 / 4  (wmma f32 16x16x4)
#define YSTRIDE 186     // even (b64-aligned), gcd(186,64)=2 -> conflict-free
#define ROWS_PER_BLOCK 16

// ---------------------------------------------------------------------------
// Pack kernel: emit basis in WMMA A-fragment order.
//   frag[tile][kk][lane] = float2{ Bb[m0][n], Bb[m0+1][n] }
//   m0 = 4*kk + 2*(lane>=16), n = tile*16 + (lane&15), tile = k*16 + i
//   Bb[m][n] = (m < 176) ? U3[n*176 + m] : U2[n*4 + (m-176)]
// ---------------------------------------------------------------------------
__global__ void sc_pack_basis(const float* __restrict__ U3,
                              const float* __restrict__ U2,
                              float2* __restrict__ P, int ntiles) {
  int total = ntiles * KSTEPS * 32;
  for (int q = blockIdx.x * blockDim.x + threadIdx.x; q < total;
       q += gridDim.x * blockDim.x) {
    int lane = q & 31;
    int kk = (q >> 5) % KSTEPS;
    int tile = q / (KSTEPS * 32);
    int m0 = 4 * kk + 2 * (lane >> 4);
    int n = tile * 16 + (lane & 15);
    float a = (m0 < 176) ? U3[n * 176 + m0] : U2[n * 4 + (m0 - 176)];
    int m1 = m0 + 1;
    float b = (m1 < 176) ? U3[n * 176 + m1] : U2[n * 4 + (m1 - 176)];
    P[q] = make_float2(a, b);
  }
}

// ---------------------------------------------------------------------------
// Main kernel: one block = 16 (b,c) rows, 8 waves cover the 9 (t,k) groups.
// ---------------------------------------------------------------------------
__global__ __launch_bounds__(256) void sc_wmma_kernel(
    const float* __restrict__ x, const int* __restrict__ indices,
    const v2f* __restrict__ P0, const v2f* __restrict__ P1,
    const v2f* __restrict__ P2,
    const float* __restrict__ U1_0, const float* __restrict__ U1_1,
    const float* __restrict__ U1_2,
    const float* __restrict__ W1_0, const float* __restrict__ W1_1,
    const float* __restrict__ W1_2,
    const float* __restrict__ W2_0, const float* __restrict__ W2_1,
    const float* __restrict__ W2_2,
    const float* __restrict__ W3_0, const float* __restrict__ W3_1,
    const float* __restrict__ W3_2,
    float* __restrict__ out) {
  __shared__ float xs[ROWS_PER_BLOCK][FEAT + 1];
  __shared__ alignas(16) float Ylds[3][ROWS_PER_BLOCK][YSTRIDE];
  __shared__ float w1s[3][ROWS_PER_BLOCK];
  __shared__ int es[ROWS_PER_BLOCK];

  const int tid = threadIdx.x;
  const int rowbase = blockIdx.x * ROWS_PER_BLOCK;

  // Stage x tile (rows = flattened (b,c)) and per-row element ids.
  {
    int r = tid >> 4, l = tid & 15;
    xs[r][l] = x[(rowbase + r) * FEAT + l];
  }
  if (tid < ROWS_PER_BLOCK) es[tid] = indices[(rowbase + tid) >> 6];
  __syncthreads();

  // Build the three per-irrep Y vectors (B-matrix operand) in LDS.
  for (int q = tid; q < 3 * ROWS_PER_BLOCK * KDIM; q += 256) {
    int t = q / (ROWS_PER_BLOCK * KDIM);
    int rm = q - t * (ROWS_PER_BLOCK * KDIM);
    int r = rm / KDIM, m = rm - r * KDIM;
    int bc = rowbase + r, c = bc & 63, e = es[r];
    const float* W3 = (t == 0) ? W3_0 : (t == 1) ? W3_1 : W3_2;
    const float* W2 = (t == 0) ? W2_0 : (t == 1) ? W2_1 : W2_2;
    float v;
    if (m < 176) {
      int l = m / 11, p = m - l * 11;
      v = xs[r][l] * W3[(e * 11 + p) * 64 + c];   // W3: (10, 11, 64)
    } else {
      v = W2[(e * 4 + (m - 176)) * 64 + c];       // W2: (10, 4, 64)
    }
    Ylds[t][r][m] = v;
  }
  if (tid < 3 * ROWS_PER_BLOCK) {
    int t = tid >> 4, r = tid & 15;
    int bc = rowbase + r, c = bc & 63;
    const float* W1 = (t == 0) ? W1_0 : (t == 1) ? W1_1 : W1_2;
    w1s[t][r] = W1[es[r] * 64 + c];               // W1: (10, 1, 64)
  }
  __syncthreads();

  const int wave = tid >> 5;
  const int lane = tid & 31;
  const int half = lane >> 4;   // 0: lanes 0-15 (j 0-7), 1: lanes 16-31 (j 8-15)
  const int l15 = lane & 15;    // my (b,c) row within the block

  // Per-lane copies of x for my row: the 8 j-values owned by my half.
  float xh[8];
#pragma unroll
  for (int v = 0; v < 8; ++v) xh[v] = xs[l15][(half << 3) + v];

  // 9 (t,k) groups: t0:k=0 | t1:k=0..2 | t2:k=0..4. Wave-uniform control flow.
  for (int g = wave; g < 9; g += 8) {
    int t, k;
    if (g < 1)       { t = 0; k = 0; }
    else if (g < 4)  { t = 1; k = g - 1; }
    else             { t = 2; k = g - 4; }
    const v2f* Pt    = (t == 0) ? P0 : (t == 1) ? P1 : P2;
    const float* U1  = (t == 0) ? U1_0 : (t == 1) ? U1_1 : U1_2;
    const int kt     = (t == 0) ? 1 : (t == 1) ? 3 : 5;
    const int seg    = (t == 0) ? 0 : (t == 1) ? 64 : 256;
    const float w1r  = w1s[t][l15];
    const float* Yrow = &Ylds[t][l15][0];

    float accT = 0.0f;                 // Horner accumulator for my row

    for (int i = 0; i < 16; ++i) {     // Horner over the last x factor
      // A-fragments for tile (k,i): packed float2 per lane per k-step.
      const v2f* Ap = Pt + ((size_t)(k * 16 + i) * KSTEPS) * 32 + lane;
      __builtin_prefetch(Ap, 0, 1);    // global_prefetch (L2-hot basis)

      v8f acc = {};                    // D = C2T tile: regs = j, lane = row
#pragma unroll 9
      for (int kk = 0; kk < KSTEPS; ++kk) {
        v2f A = Ap[kk * 32];           // coalesced global_load_b64
        v2f B = *reinterpret_cast<const v2f*>(Yrow + 4 * kk + 2 * half);
        acc = __builtin_amdgcn_wmma_f32_16x16x4_f32(
            false, A, false, B, (short)0, acc, false, false);
      }

      // j-contraction: regs hold j = v + 8*half for my row; sum in registers,
      // then one cross-half combine.
      float s = 0.0f;
#pragma unroll
      for (int v = 0; v < 8; ++v) s = fmaf(acc[v], xh[v], s);
      s += __shfl_xor(s, 16, 32);

      // Degree-1 basis + Horner step.
      const float u1v = U1[k * 16 + i];    // U1_t[k][i][0]
      const float c1 = fmaf(u1v, w1r, s);
      accT = fmaf(c1, xs[l15][i], accT);
    }

    // Half 0 writes row l15 of this (t,k) output.
    if (lane < 16) {
      const int bc = rowbase + l15;
      const int b = bc >> 6, c = bc & 63;
      out[b * 576 + seg + c * kt + k] = accT;
    }
  }
}

// ---------------------------------------------------------------------------
extern "C" void kernel_launch(void* const* d_in, const int* in_sizes, int n_in,
                              void* d_out, int out_size, void* d_ws, size_t ws_size,
                              hipStream_t stream) {
  (void)in_sizes; (void)n_in; (void)out_size; (void)ws_size;
  const float* x   = (const float*)d_in[0];
  const int* idx   = (const int*)d_in[1];
  const float* U1_0 = (const float*)d_in[2];
  const float* W1_0 = (const float*)d_in[3];
  const float* U2_0 = (const float*)d_in[4];
  const float* W2_0 = (const float*)d_in[5];
  const float* U3_0 = (const float*)d_in[6];
  const float* W3_0 = (const float*)d_in[7];
  const float* U1_1 = (const float*)d_in[8];
  const float* W1_1 = (const float*)d_in[9];
  const float* U2_1 = (const float*)d_in[10];
  const float* W2_1 = (const float*)d_in[11];
  const float* U3_1 = (const float*)d_in[12];
  const float* W3_1 = (const float*)d_in[13];
  const float* U1_2 = (const float*)d_in[14];
  const float* W1_2 = (const float*)d_in[15];
  const float* U2_2 = (const float*)d_in[16];
  const float* W2_2 = (const float*)d_in[17];
  const float* U3_2 = (const float*)d_in[18];
  const float* W3_2 = (const float*)d_in[19];
  float* out = (float*)d_out;

  // Workspace: basis in WMMA-fragment order, float2 per lane per k-step.
  // t0: 16 tiles, t1: 48 tiles, t2: 80 tiles; tile = 45*32 float2 (1.66 MB).
  float2* P0 = (float2*)d_ws;
  float2* P1 = P0 + (size_t)16 * KSTEPS * 32;
  float2* P2 = P1 + (size_t)48 * KSTEPS * 32;

  sc_pack_basis<<<64, 256, 0, stream>>>(U3_0, U2_0, P0, 16);
  sc_pack_basis<<<128, 256, 0, stream>>>(U3_1, U2_1, P1, 48);
  sc_pack_basis<<<256, 256, 0, stream>>>(U3_2, U2_2, P2, 80);

  // 16384 (b,c) rows / 16 rows per block = 1024 blocks, 256 threads (8 waves).
  sc_wmma_kernel<<<1024, 256, 0, stream>>>(
      x, idx, (const v2f*)P0, (const v2f*)P1, (const v2f*)P2,
      U1_0, U1_1, U1_2,
      W1_0, W1_1, W1_2,
      W2_0, W2_1, W2_2,
      W3_0, W3_1, W3_2,
      out);
}